// DFFN_57836029608181
// MI455X (gfx1250) — compile-verified
//
#include <hip/hip_runtime.h>
#include <math.h>

typedef float v2f __attribute__((ext_vector_type(2)));
typedef float v4f __attribute__((ext_vector_type(4)));
typedef float v8f __attribute__((ext_vector_type(8)));
typedef unsigned int  u32;
typedef unsigned long long u64;
typedef u32 u32x4 __attribute__((ext_vector_type(4)));
typedef int i32x4 __attribute__((ext_vector_type(4)));
typedef int i32x8 __attribute__((ext_vector_type(8)));

#define HD 256
#define WD 256
#define HW 65536

// cos(2*pi*i/8), sin(2*pi*i/8)
__device__ constexpr float CT8[8] = { 1.f,  0.70710678118654752f,  0.f, -0.70710678118654752f,
                                     -1.f, -0.70710678118654752f,  0.f,  0.70710678118654752f };
__device__ constexpr float ST8[8] = { 0.f,  0.70710678118654752f,  1.f,  0.70710678118654752f,
                                      0.f, -0.70710678118654752f, -1.f, -0.70710678118654752f };

#if __has_builtin(__builtin_amdgcn_tensor_load_to_lds)
#define USE_TDM 1
#else
#define USE_TDM 0
#endif

#if USE_TDM
// Build a Tensor DMA Descriptor (CDNA5 ISA D# layout) and launch
// TENSOR_LOAD_TO_LDS for a <=3D tile of f32, packed into LDS with
// per-line padding (pad inserted after every 2^(pi+1) DWORDs, pa+1 DWORDs).
// tile_d2==0 -> 2D tile. Strides in elements (f32).
// Toolchain probe: 6-arg builtin (u32x4, i32x8, i32x4, i32x4, i32x8, i32).
__device__ __forceinline__ void tdm_load_tile_f32(
    u32 lds_off, const float* gptr,
    u32 tile_d0, u32 tile_d1, u32 tile_d2,
    u64 stride0, u64 stride1,
    u32 pad_interval, u32 pad_amount_m1)
{
    const u64 ga = (u64)(uintptr_t)gptr;
    const u64 s0b = stride0;            // tensor_dim0_stride, data_size units
    const u64 s1b = stride1;            // tensor_dim1_stride
    const u32 T   = 0x40000000u;        // huge tensor dims -> OOB never clips

    u32x4 g0;
    g0[0] = 1u;                                          // count=1, user mode
    g0[1] = lds_off;                                     // LDS byte address
    g0[2] = (u32)ga;                                     // global_addr[31:0]
    g0[3] = (u32)((ga >> 32) & 0x1FFFFFFu) | (2u << 30); // [56:32] + type=2

    i32x8 g1;
    g1[0] = (int)((2u << 16)                 // data_size = 4B
                | (1u << 20)                 // pad_enable
                | (pad_interval << 22)       // pad every 2^(pi+1) DWORDs
                | (pad_amount_m1 << 25));    // pad (pa+1) DWORDs
    g1[1] = (int)((T & 0xFFFFu) << 16);                  // tensor_dim0[15:0]
    g1[2] = (int)((T >> 16) | ((T & 0xFFFFu) << 16));    // dim0 hi | dim1 lo
    g1[3] = (int)((T >> 16) | (tile_d0 << 16));          // dim1 hi | tile_dim0
    g1[4] = (int)(tile_d1 | (tile_d2 << 16));            // tile_dim1 | tile_dim2
    g1[5] = (int)(u32)s0b;                               // dim0_stride[31:0]
    g1[6] = (int)((u32)(s0b >> 32) | ((u32)(s1b & 0xFFFFu) << 16));
    g1[7] = (int)(u32)(s1b >> 16);                       // dim1_stride[47:16]

    i32x4 g2;
    g2[0] = (int)T;                                      // tensor_dim2
    g2[1] = (int)T;                                      // tensor_dim3 (unused)
    g2[2] = 0;                                           // dim2_stride lo (unused: tile_dim3=0)
    g2[3] = 0;                                           // dim2_stride hi | tile_dim3=0
    i32x4 g3 = {0, 0, 0, 0};                             // dims 3/4 unused
    i32x8 gz = {0, 0, 0, 0, 0, 0, 0, 0};                 // 5th operand: zero-filled

    __builtin_amdgcn_tensor_load_to_lds(g0, g1, g2, g3, gz, 0);
}
#endif

// ---------------------------------------------------------------------------
// Kernel 1: project_in (64 -> 256, fp32 WMMA) fused with per-8x8-patch
//           rfft2 * fft_params * irfft2 (exact, FFTW C2R semantics).
// Workgroup: 256 threads (8 waves). Spatial tile: 8 rows x 32 cols of one batch.
// ---------------------------------------------------------------------------
__global__ __launch_bounds__(256) void k_proj_in_fft(
    const float* __restrict__ x, const float* __restrict__ w_in,
    const float* __restrict__ fftp, float* __restrict__ hf)
{
    __shared__ float x_lds[64][260];   // 64 in-channels x 256 px (+4 DW pad)
    __shared__ float h_lds[64][260];   // one 64-channel chunk of h

    const int t  = threadIdx.x;
    const int b  = blockIdx.x >> 8;          // 8 batches
    const int pr = (blockIdx.x >> 3) & 31;   // patch-row (8 rows each)
    const int cb = blockIdx.x & 7;           // col-block (32 cols each)
    const int r0 = pr * 8, c0 = cb * 32;

    __builtin_prefetch(w_in, 0, 3);          // global_prefetch_b8: weights -> cache

    const int lane = t & 31, w = t >> 5;

#if USE_TDM
    // TDM DMA: 3D tile (32 cols x 8 rows x 64 ch), per-channel 256 DWORDs
    // followed by a 4-DWORD pad -> exactly the [64][260] LDS layout.
    if (w == 0) {
        tdm_load_tile_f32((u32)(uintptr_t)&x_lds[0][0],
                          x + ((size_t)(b * 64) * HD + r0) * WD + c0,
                          /*d0=*/32, /*d1=*/8, /*d2=*/64,
                          /*s0=*/WD, /*s1=*/HW,
                          /*pad_interval=*/7 /*256 DW*/, /*pad_amount-1=*/3 /*4 DW*/);
        __builtin_amdgcn_s_wait_tensorcnt(0);
    }
#else
    for (int i = 0; i < 64; ++i) {
        int idx = t + i * 256;
        int c = idx >> 8, p = idx & 255;
        int row = p >> 5, col = p & 31;
        x_lds[c][p] = x[((size_t)(b * 64 + c) * HD + (r0 + row)) * WD + c0 + col];
    }
#endif
    __syncthreads();

    const int o_tile = w >> 1, p_grp = w & 1;   // 4 o-tiles x 2 p-groups
    const int lhi = lane >> 4, llo = lane & 15;

    #pragma unroll 1
    for (int chunk = 0; chunk < 4; ++chunk) {   // 4 chunks of 64 out-channels
        // ---- GEMM: h[o, p] = sum_c w_in[o,c] * x[c,p] ----
        const int o_row = chunk * 64 + o_tile * 16 + llo;
        v2f a[16];
        #pragma unroll
        for (int ks = 0; ks < 16; ++ks)         // A frag: M=llo, K=ks*4+2*lhi+{0,1}
            a[ks] = *(const v2f*)(w_in + o_row * 64 + ks * 4 + (lhi << 1));

        #pragma unroll 1
        for (int sub = 0; sub < 8; ++sub) {     // 8 sub-tiles of 16 px -> 128 px/wave
            const int p_l = p_grp * 128 + sub * 16 + llo;
            v8f acc = {};
            #pragma unroll
            for (int ks = 0; ks < 16; ++ks) {
                const int k = ks * 4 + (lhi << 1);
                v2f bf = { x_lds[k][p_l], x_lds[k + 1][p_l] };
                acc = __builtin_amdgcn_wmma_f32_16x16x4_f32(
                          false, a[ks], false, bf, (short)0, acc, false, false);
            }
            #pragma unroll
            for (int r = 0; r < 8; ++r)         // D: M = r + 8*lhi
                h_lds[o_tile * 16 + r + (lhi << 3)][p_l] = acc[r];
        }
        __syncthreads();

        // ---- FFT filter: one thread per (channel-in-chunk, 8x8 patch) ----
        {
            const int ch = t >> 2, pt = t & 3;
            const int cg = chunk * 64 + ch;
            const float* sp = fftp + cg * 40;   // (8 ky) x (5 kx) real scales

            float Ar[8][5], Ai[8][5];
            #pragma unroll
            for (int y = 0; y < 8; ++y) {       // rfft along x per row
                float rv[8];
                #pragma unroll
                for (int xx = 0; xx < 8; ++xx) rv[xx] = h_lds[ch][y * 32 + pt * 8 + xx];
                #pragma unroll
                for (int k = 0; k < 5; ++k) {
                    float ar = 0.f, ai = 0.f;
                    #pragma unroll
                    for (int xx = 0; xx < 8; ++xx) {
                        const int ix = (xx * k) & 7;
                        ar += rv[xx] * CT8[ix];
                        ai -= rv[xx] * ST8[ix];
                    }
                    Ar[y][k] = ar; Ai[y][k] = ai;
                }
            }
            float Br[8][5], Bi[8][5];
            #pragma unroll
            for (int ky = 0; ky < 8; ++ky) {    // full fft along y, then scale
                #pragma unroll
                for (int k = 0; k < 5; ++k) {
                    float br = 0.f, bi = 0.f;
                    #pragma unroll
                    for (int y = 0; y < 8; ++y) {
                        const int iy = (y * ky) & 7;
                        br += Ar[y][k] * CT8[iy] + Ai[y][k] * ST8[iy];
                        bi += Ai[y][k] * CT8[iy] - Ar[y][k] * ST8[iy];
                    }
                    const float s = sp[ky * 5 + k];
                    Br[ky][k] = br * s; Bi[ky][k] = bi * s;
                }
            }
            #pragma unroll
            for (int y = 0; y < 8; ++y) {       // ifft along y, irfft along x
                float Cr[5], Ci[5];
                #pragma unroll
                for (int k = 0; k < 5; ++k) {
                    float cr = 0.f, ci = 0.f;
                    #pragma unroll
                    for (int ky = 0; ky < 8; ++ky) {
                        const int iy = (y * ky) & 7;
                        cr += Br[ky][k] * CT8[iy] - Bi[ky][k] * ST8[iy];
                        ci += Bi[ky][k] * CT8[iy] + Br[ky][k] * ST8[iy];
                    }
                    Cr[k] = cr; Ci[k] = ci;
                }
                #pragma unroll
                for (int xx = 0; xx < 8; ++xx) {
                    float o = 0.f;
                    #pragma unroll
                    for (int k = 0; k < 5; ++k) {
                        const int ix = (xx * k) & 7;
                        const float wgt = (k == 0 || k == 4) ? 1.f : 2.f; // Hermitian fold
                        o += wgt * (Cr[k] * CT8[ix] - Ci[k] * ST8[ix]);   // ST=0 at k=0,4
                    }
                    hf[((size_t)(b * 256 + cg) * HD + (r0 + y)) * WD + c0 + pt * 8 + xx]
                        = o * (1.f / 64.f);
                }
            }
        }
        __syncthreads();
    }
}

// ---------------------------------------------------------------------------
// Kernel 2: depthwise 3x3 (SAME, zero pad) + exact-GELU gate: g = gelu(d1)*d2
// 4 pixels per thread; neighborhood loads shared across the 4 outputs.
// ---------------------------------------------------------------------------
__global__ __launch_bounds__(256) void k_dw_gate(
    const float* __restrict__ hf, const float* __restrict__ wdw,
    float* __restrict__ g)
{
    const size_t idx = (size_t)blockIdx.x * 256 + threadIdx.x;  // B*128*HW/4 groups
    const int q   = (int)(idx & 16383);         // 4-px group within image plane
    const int c   = (int)((idx >> 14) & 127);
    const int b   = (int)(idx >> 21);
    const int px0 = q * 4;
    const int y = px0 >> 8, x0 = px0 & 255;     // x0 in {0,4,...,252}

    const float* w1 = wdw + c * 9;
    const float* w2 = wdw + (c + 128) * 9;
    const float* p1 = hf + (size_t)(b * 256 + c) * HW;
    const float* p2 = hf + (size_t)(b * 256 + c + 128) * HW;

    // load 3 rows x 6 cols neighborhood for both channels (zero padded)
    float n1[3][6], n2[3][6];
    #pragma unroll
    for (int dy = 0; dy < 3; ++dy) {
        const int yy = y + dy - 1;
        const bool yok = (yy >= 0) && (yy <= 255);
        #pragma unroll
        for (int dx = 0; dx < 6; ++dx) {
            const int xx = x0 + dx - 1;
            const bool ok = yok && (xx >= 0) && (xx <= 255);
            const int off = yy * 256 + xx;
            n1[dy][dx] = ok ? p1[off] : 0.f;
            n2[dy][dx] = ok ? p2[off] : 0.f;
        }
    }
    v4f res;
    #pragma unroll
    for (int j = 0; j < 4; ++j) {
        float d1 = 0.f, d2 = 0.f;
        #pragma unroll
        for (int dy = 0; dy < 3; ++dy)
            #pragma unroll
            for (int dx = 0; dx < 3; ++dx) {
                d1 += w1[dy * 3 + dx] * n1[dy][j + dx];
                d2 += w2[dy * 3 + dx] * n2[dy][j + dx];
            }
        const float ge = 0.5f * d1 * (1.f + erff(d1 * 0.70710678118654752f));
        res[j] = ge * d2;
    }
    *(v4f*)(g + ((size_t)(b * 128 + c) << 16) + px0) = res;   // 16B-aligned store
}

// ---------------------------------------------------------------------------
// Kernel 3: project_out (128 -> 64, fp32 WMMA). 256 threads; 128-px tile.
// ---------------------------------------------------------------------------
__global__ __launch_bounds__(256) void k_proj_out(
    const float* __restrict__ g, const float* __restrict__ w_out,
    float* __restrict__ out)
{
    __shared__ float g_lds[128][132];
    const int t = threadIdx.x;
    const size_t base = (size_t)blockIdx.x * 128;
    const int b   = (int)(base >> 16);
    const int px0 = (int)(base & 65535);

    __builtin_prefetch(w_out, 0, 3);

    const int lane = t & 31, w = t >> 5;

#if USE_TDM
    // TDM DMA: 2D tile (128 px x 128 ch), per-channel 128 DWORDs + 4 DW pad
    // -> the [128][132] LDS layout.
    if (w == 0) {
        tdm_load_tile_f32((u32)(uintptr_t)&g_lds[0][0],
                          g + ((size_t)(b * 128) << 16) + px0,
                          /*d0=*/128, /*d1=*/128, /*d2=*/0,
                          /*s0=*/HW, /*s1=*/0,
                          /*pad_interval=*/6 /*128 DW*/, /*pad_amount-1=*/3 /*4 DW*/);
        __builtin_amdgcn_s_wait_tensorcnt(0);
    }
#else
    for (int i = 0; i < 64; ++i) {           // stage g tile: 128 ch x 128 px
        int idx = t + i * 256;
        int c = idx >> 7, p = idx & 127;
        g_lds[c][p] = g[((size_t)(b * 128 + c) << 16) + px0 + p];
    }
#endif
    __syncthreads();

    const int o_tile = w >> 1, p_grp = w & 1;    // 4 o-tiles x 2 p-groups
    const int lhi = lane >> 4, llo = lane & 15;

    const int o_row = o_tile * 16 + llo;
    v2f a[32];
    #pragma unroll
    for (int ks = 0; ks < 32; ++ks)
        a[ks] = *(const v2f*)(w_out + o_row * 128 + ks * 4 + (lhi << 1));

    #pragma unroll 1
    for (int sub = 0; sub < 4; ++sub) {
        const int p_l = p_grp * 64 + sub * 16 + llo;
        v8f acc = {};
        #pragma unroll
        for (int ks = 0; ks < 32; ++ks) {
            const int k = ks * 4 + (lhi << 1);
            v2f bf = { g_lds[k][p_l], g_lds[k + 1][p_l] };
            acc = __builtin_amdgcn_wmma_f32_16x16x4_f32(
                      false, a[ks], false, bf, (short)0, acc, false, false);
        }
        #pragma unroll
        for (int r = 0; r < 8; ++r)
            out[((size_t)(b * 64 + o_tile * 16 + r + (lhi << 3)) << 16) + px0 + p_l] = acc[r];
    }
}

// ---------------------------------------------------------------------------
extern "C" void kernel_launch(void* const* d_in, const int* in_sizes, int n_in,
                              void* d_out, int out_size, void* d_ws, size_t ws_size,
                              hipStream_t stream)
{
    const float* x     = (const float*)d_in[0];   // (8, 64, 256, 256)
    const float* w_in  = (const float*)d_in[1];   // (256, 64)
    const float* w_dw  = (const float*)d_in[2];   // (256, 1, 3, 3)
    const float* fftp  = (const float*)d_in[3];   // (256, 1, 1, 8, 5)
    const float* w_out = (const float*)d_in[4];   // (64, 128)
    float* out = (float*)d_out;                   // (8, 64, 256, 256)

    float* hf = (float*)d_ws;                         // 8*256*65536 f32 = 512 MB
    float* gb = hf + (size_t)8 * 256 * HW;            // 8*128*65536 f32 = 256 MB
    (void)in_sizes; (void)n_in; (void)out_size; (void)ws_size;

    k_proj_in_fft<<<dim3(8 * 32 * 8), dim3(256), 0, stream>>>(x, w_in, fftp, hf);
    k_dw_gate   <<<dim3((8 * 128 * HW) / (256 * 4)), dim3(256), 0, stream>>>(hf, w_dw, gb);
    k_proj_out  <<<dim3((8 * HW) / 128), dim3(256), 0, stream>>>(gb, w_out, out);
}